// NonMaxSuppression_8890582303353
// MI455X (gfx1250) — compile-verified
//
#include <hip/hip_runtime.h>

// NonMaxSuppression for B=1, C=80, N=1024, IOU_T=0.5, SCORE_T=0.5, MAX_OUT=100.
// Strategy (latency-bound problem):
//  kernel 1: one 1024-thread workgroup per class.
//    - async-copy boxes (16KB) global -> LDS (gfx1250 ASYNCcnt path)
//    - bitonic-sort 64-bit keys (~score_bits<<32 | idx) ascending
//      == stable argsort of -scores
//    - gather sorted boxes + areas into LDS
//    - compute IoU suppression bitmask in 128-row chunks (16KB LDS),
//      greedy scan on wave 0 with lane-owned 32-bit removed words
//  kernel 2: single block: prefix-sum per-class counts, zero output,
//    write compacted (0, c, idx) rows.

#define NMS_N      1024
#define NMS_MAXOUT 100
#define NMS_CHUNK  128
#define NMS_NCHUNK (NMS_N / NMS_CHUNK)

typedef unsigned long long u64;
typedef unsigned int       u32;
typedef int v4i __attribute__((vector_size(16)));   // matches builtin param pointee

struct __align__(16) Box4 { float y1, x1, y2, x2; };

#if __has_builtin(__builtin_amdgcn_global_load_async_to_lds_b128) && \
    __has_builtin(__builtin_amdgcn_s_wait_asynccnt)
#define NMS_USE_ASYNC 1
#endif

__global__ __launch_bounds__(1024) void nms_class_kernel(
    const float* __restrict__ boxes,    // [N,4]  (y1,x1,y2,x2)
    const float* __restrict__ scores,   // [C,N]
    int* __restrict__ ws_idx,           // [C, MAX_OUT]
    int* __restrict__ ws_cnt)           // [C]
{
  __shared__ u64  keys[NMS_N];              //  8 KB
  __shared__ Box4 lbox[NMS_N];              // 16 KB (unsorted boxes)
  __shared__ Box4 sbox[NMS_N];              // 16 KB (sorted boxes)
  __shared__ float sarea[NMS_N];            //  4 KB
  __shared__ u32  mask[NMS_CHUNK][32];      // 16 KB (IoU>T bitmask, chunked)

  const int tid  = threadIdx.x;
  const int lane = tid & 31;
  const int c    = blockIdx.x;

  // ---- stage 0: boxes -> LDS via CDNA5 async load (ASYNCcnt) ----
#ifdef NMS_USE_ASYNC
  __builtin_amdgcn_global_load_async_to_lds_b128(
      (v4i*)(boxes + (size_t)tid * 4), (v4i*)&lbox[tid], 0, 0);
#else
  lbox[tid] = ((const Box4*)boxes)[tid];
#endif

  // ---- stage 1: sort keys ascending == scores descending, idx tiebreak ----
  // scores in [0,1): float bits are monotone, so key hi = ~bits(score).
  {
    u32 sb = __float_as_uint(scores[(size_t)c * NMS_N + tid]);
    keys[tid] = ((u64)(~sb) << 32) | (u32)tid;
  }
  for (int k = 2; k <= NMS_N; k <<= 1) {
    for (int j = k >> 1; j > 0; j >>= 1) {
      __syncthreads();
      int ixj = tid ^ j;
      if (ixj > tid) {
        u64 a = keys[tid], b = keys[ixj];
        bool doswap = ((tid & k) == 0) ? (a > b) : (a < b);
        if (doswap) { keys[tid] = b; keys[ixj] = a; }
      }
    }
  }
  __syncthreads();

  // ---- stage 2: gather sorted boxes + areas ----
#ifdef NMS_USE_ASYNC
  __builtin_amdgcn_s_wait_asynccnt(0);   // this wave's async loads landed
  __syncthreads();                       // every wave's loads landed
#endif
  {
    u32 si = (u32)(keys[tid] & 0xffffffffu);
    Box4 bb = lbox[si];
    sbox[tid]  = bb;
    sarea[tid] = (bb.y2 - bb.y1) * (bb.x2 - bb.x1);
  }
  __syncthreads();

  // ---- stage 3: chunked IoU bitmask + wave32 greedy scan ----
  u32 removed = 0;        // lane l owns suppression bits for j in [32l, 32l+32)
  int outcount = 0;       // meaningful on wave 0 only
  for (int ch = 0; ch < NMS_NCHUNK; ++ch) {
    const int r0 = ch * NMS_CHUNK;
    // all 1024 threads: build mask rows r0..r0+127 (4 words per thread)
    for (int t = tid; t < NMS_CHUNK * 32; t += 1024) {
      const int r = t >> 5, w = t & 31;
      const int i = r0 + r;
      Box4 bi = sbox[i];
      float ai = sarea[i];
      u32 bits = 0;
      for (int jj = 0; jj < 32; ++jj) {
        int j = (w << 5) + jj;
        Box4 bj = sbox[j];
        float ih = fminf(bi.y2, bj.y2) - fmaxf(bi.y1, bj.y1);
        ih = ih > 0.0f ? ih : 0.0f;
        float iw = fminf(bi.x2, bj.x2) - fmaxf(bi.x1, bj.x1);
        iw = iw > 0.0f ? iw : 0.0f;
        float inter = ih * iw;
        float iou = inter / (ai + sarea[j] - inter);
        bits |= (iou > 0.5f ? 1u : 0u) << jj;
      }
      mask[r][w] = bits;
    }
    __syncthreads();
    // wave 0: greedy scan over this chunk's rows
    if (tid < 32) {
      for (int r = 0; r < NMS_CHUNK; ++r) {
        const int i = r0 + r;
        u64 key = keys[i];                               // uniform broadcast
        float sc = __uint_as_float(~(u32)(key >> 32));
        u32 rw = __shfl(removed, i >> 5);                // owner lane's word
        bool rem = (rw >> (i & 31)) & 1u;
        if (!rem && sc > 0.5f) {                         // kept
          if (lane == 0 && outcount < NMS_MAXOUT)
            ws_idx[c * NMS_MAXOUT + outcount] = (int)(u32)(key & 0xffffffffu);
          outcount++;
          removed |= mask[r][lane];
        }
      }
    }
    __syncthreads();
  }
  if (tid == 0)
    ws_cnt[c] = outcount < NMS_MAXOUT ? outcount : NMS_MAXOUT;
}

__global__ __launch_bounds__(256) void nms_compact_kernel(
    const int* __restrict__ ws_idx,     // [C, MAX_OUT]
    const int* __restrict__ ws_cnt,     // [C]
    int* __restrict__ out,              // [C*MAX_OUT, 3]
    int C)
{
  __shared__ int offs[256];
  const int tid = threadIdx.x;
  if (tid == 0) {
    int acc = 0;
    for (int cc = 0; cc < C; ++cc) { offs[cc] = acc; acc += ws_cnt[cc]; }
  }
  const int total_words = C * NMS_MAXOUT * 3;
  for (int i = tid; i < total_words; i += 256) out[i] = 0;
  __syncthreads();
  for (int t = tid; t < C * NMS_MAXOUT; t += 256) {
    int cc = t / NMS_MAXOUT;
    int k  = t - cc * NMS_MAXOUT;
    if (k < ws_cnt[cc]) {
      int p = (offs[cc] + k) * 3;
      out[p + 0] = 0;             // b == 0 (B = 1)
      out[p + 1] = cc;
      out[p + 2] = ws_idx[t];
    }
  }
}

extern "C" void kernel_launch(void* const* d_in, const int* in_sizes, int n_in,
                              void* d_out, int out_size, void* d_ws, size_t ws_size,
                              hipStream_t stream) {
  (void)n_in; (void)out_size; (void)ws_size;
  const float* boxes  = (const float*)d_in[0];   // (1, 1024, 4) f32
  const float* scores = (const float*)d_in[1];   // (1, 80, 1024) f32
  const int N = NMS_N;
  const int C = in_sizes[1] / N;                 // 80

  int* ws_idx = (int*)d_ws;                      // C*MAX_OUT ints
  int* ws_cnt = ws_idx + C * NMS_MAXOUT;         // C ints
  int* out    = (int*)d_out;                     // C*MAX_OUT*3 int32

  nms_class_kernel<<<C, 1024, 0, stream>>>(boxes, scores, ws_idx, ws_cnt);
  nms_compact_kernel<<<1, 256, 0, stream>>>(ws_idx, ws_cnt, out, C);
}